// LikelihoodRatioTest_62362925138760
// MI455X (gfx1250) — compile-verified
//
#include <hip/hip_runtime.h>

typedef __attribute__((ext_vector_type(2))) float v2f;
typedef __attribute__((ext_vector_type(8))) float v8f;

#define CDIM 64
#define RETRO_EPOCH 10

__global__ void lrt_zero_acc(float* acc) {
    acc[0] = 0.0f;
    acc[1] = 0.0f;
}

__launch_bounds__(256)
__global__ void lrt_main_kernel(const float* __restrict__ outputs,
                                const float* __restrict__ targets,
                                const float* __restrict__ A,
                                const float* __restrict__ soft_labels,
                                const int*   __restrict__ index,
                                float* __restrict__ acc,
                                int B) {
    // Per-wave LDS slab: 64 floats of s, 64 floats of ls. 8 waves -> 4 KB.
    __shared__ float lds[8 * 2 * CDIM];

    const int lane = threadIdx.x & 31;
    const int wave = threadIdx.x >> 5;
    const int b = blockIdx.x * 8 + wave;
    if (b >= B) return;   // wave-uniform: full waves keep EXEC all-1s for WMMA

    float* s_sm  = &lds[wave * 2 * CDIM];
    float* ls_sm = s_sm + CDIM;

    // ---------------- log_softmax over the 64 classes of row b ----------------
    const float* orow = outputs + (size_t)b * CDIM;
    const float* trow = targets + (size_t)b * CDIM;
    float x0 = orow[lane];
    float x1 = orow[lane + 32];
    float mx = fmaxf(x0, x1);
    #pragma unroll
    for (int off = 16; off >= 1; off >>= 1)
        mx = fmaxf(mx, __shfl_xor(mx, off, 32));
    float ex = __expf(x0 - mx) + __expf(x1 - mx);
    #pragma unroll
    for (int off = 16; off >= 1; off >>= 1)
        ex += __shfl_xor(ex, off, 32);
    float lse = mx + __logf(ex);
    float ls0 = x0 - lse;
    float ls1 = x1 - lse;

    // CE partial: sum(ls * targets) for this row (negated/scaled in finalize)
    float ce_lane = ls0 * trow[lane] + ls1 * trow[lane + 32];

    // ---------------- gather per-sample A tile and soft label ----------------
    const int idx = index[b];
    const float* Ab   = A + (size_t)idx * (CDIM * CDIM);
    const float* srow = soft_labels + (size_t)idx * CDIM;

    s_sm[lane]       = srow[lane];
    s_sm[lane + 32]  = srow[lane + 32];
    ls_sm[lane]      = ls0;
    ls_sm[lane + 32] = ls1;
    // DS ops are in-order within a wave; wait + compiler memory fence for safety
    asm volatile("s_wait_dscnt 0" ::: "memory");

    // ---------------- bilinear form ls^T * Ab * s via V_WMMA_F32_16X16X4 -----
    // A-operand (16x4 f32, 2 VGPRs): lane M = lane&15; VGPR0/1 hold K = 2*khalf,
    // 2*khalf+1  => one b64 load of 2 consecutive row elements per lane.
    // B-operand (4x16 f32, 2 VGPRs): B[k][n] = s[base+k] broadcast over N.
    // D columns all replicate v = Ab_tile @ s  (16x redundancy, compute is free).
    const int mrow  = lane & 15;
    const int khalf = lane >> 4;
    float lrt_lane = 0.0f;

    #pragma unroll
    for (int p = 0; p < 4; ++p) {
        v8f dacc = {};
        #pragma unroll
        for (int q = 0; q < 4; ++q) {
            #pragma unroll
            for (int c = 0; c < 4; ++c) {
                const int base = 16 * q + 4 * c;
                v2f a = *(const v2f*)(Ab + (size_t)(16 * p + mrow) * CDIM
                                         + base + 2 * khalf);
                v2f bb;
                bb.x = s_sm[base + 2 * khalf];
                bb.y = s_sm[base + 2 * khalf + 1];
                dacc = __builtin_amdgcn_wmma_f32_16x16x4_f32(
                    /*neg_a=*/false, a, /*neg_b=*/false, bb,
                    /*c_mod=*/(short)0, dacc,
                    /*reuse_a=*/false, /*reuse_b=*/false);
            }
        }
        // C/D layout: VGPR r, lanes 0-15 -> M=r, lanes 16-31 -> M=r+8
        #pragma unroll
        for (int r = 0; r < 8; ++r)
            lrt_lane += dacc[r] * ls_sm[16 * p + r + 8 * khalf];
    }

    // ---------------- wave reductions + global accumulation ----------------
    #pragma unroll
    for (int off = 16; off >= 1; off >>= 1) {
        ce_lane  += __shfl_xor(ce_lane,  off, 32);
        lrt_lane += __shfl_xor(lrt_lane, off, 32);
    }
    if (lane == 0) {
        atomicAdd(&acc[0], ce_lane);
        atomicAdd(&acc[1], lrt_lane);   // carries a 16x duplication factor
    }
}

__global__ void lrt_finalize(const float* __restrict__ acc,
                             const int*   __restrict__ epoch,
                             float* __restrict__ out, int B) {
    float ce  = -acc[0] / (float)B;
    float res = ce;
    if (epoch[0] >= RETRO_EPOCH)
        res += -acc[1] / (16.0f * (float)B);   // undo 16x column duplication
    out[0] = res;
}

extern "C" void kernel_launch(void* const* d_in, const int* in_sizes, int n_in,
                              void* d_out, int out_size, void* d_ws, size_t ws_size,
                              hipStream_t stream) {
    // setup_inputs order: outputs, targets, A, soft_labels, epoch, index
    const float* outputs     = (const float*)d_in[0];
    const float* targets     = (const float*)d_in[1];
    const float* A           = (const float*)d_in[2];
    const float* soft_labels = (const float*)d_in[3];
    const int*   epoch       = (const int*)d_in[4];
    const int*   index       = (const int*)d_in[5];

    const int B = in_sizes[0] / CDIM;       // 8192
    float* acc = (float*)d_ws;              // 2 fp32 accumulators

    lrt_zero_acc<<<1, 1, 0, stream>>>(acc);

    const int blocks = (B + 7) / 8;         // 8 waves (samples) per 256-thread block
    lrt_main_kernel<<<blocks, 256, 0, stream>>>(outputs, targets, A, soft_labels,
                                                index, acc, B);

    lrt_finalize<<<1, 1, 0, stream>>>(acc, epoch, (float*)d_out, B);
}